// LSTM_3590592660256
// MI455X (gfx1250) — compile-verified
//
#include <hip/hip_runtime.h>
#include <cmath>

// ---------------------------------------------------------------------------
// LSTM forward for MI455X (gfx1250), f32 WMMA (V_WMMA_F32_16X16X4_F32).
//
//  - pack_weights: weights stored in WMMA B-fragment order:
//        Wpk[(k>>1)*4096 + col*2 + (k&1)] = [W|U]_gate[k][n],  col = g*512+n
//    so each lane's (K, K+1) pair is one 8-byte load, and the 4 gate columns
//    are fixed immediate offsets from one base pointer.
//  - pack_bias:    bias[2048]
//  - lstm_all:     8 independent workgroups (16 batch rows each, full hidden
//                  state). All T=512 steps inside one kernel launch; h lives
//                  in LDS (16x516 padded, bank-conflict free), c in VGPRs.
// ---------------------------------------------------------------------------

typedef __attribute__((ext_vector_type(2))) float v2f;
typedef __attribute__((ext_vector_type(8))) float v8f;

#define B_   128
#define T_   512
#define F_   256
#define H_   512
#define KTOT 768           // F_ + H_
#define NCOL 2048          // 4 * H_
#define WC_FLOATS (KTOT * NCOL)

__device__ __forceinline__ v8f wmma_f32(v2f a, v2f b, v8f c) {
  // (neg_a, A, neg_b, B, c_mod, C, reuse_a, reuse_b)
  return __builtin_amdgcn_wmma_f32_16x16x4_f32(false, a, false, b, (short)0, c,
                                               false, false);
}

__device__ __forceinline__ float sigmoidf_(float x) {
  return 1.0f / (1.0f + __expf(-x));
}

__global__ void pack_weights(const float* __restrict__ Wi, const float* __restrict__ Ui,
                             const float* __restrict__ Wf, const float* __restrict__ Uf,
                             const float* __restrict__ Wc, const float* __restrict__ Uc,
                             const float* __restrict__ Wo, const float* __restrict__ Uo,
                             float* __restrict__ Wpk) {
  int tid = blockIdx.x * blockDim.x + threadIdx.x;
  if (tid >= WC_FLOATS) return;
  int k   = tid / NCOL;
  int col = tid - k * NCOL;
  int g   = col >> 9;        // gate 0..3  (i, f, c, o)
  int n   = col & (H_ - 1);  // column within H
  const float* Wg = (g == 0) ? Wi : (g == 1) ? Wf : (g == 2) ? Wc : Wo;
  const float* Ug = (g == 0) ? Ui : (g == 1) ? Uf : (g == 2) ? Uc : Uo;
  float v = (k < F_) ? Wg[k * H_ + n] : Ug[(k - F_) * H_ + n];
  // WMMA B-fragment order: pair-interleave consecutive K rows.
  Wpk[(size_t)(k >> 1) * (2 * NCOL) + (size_t)col * 2 + (k & 1)] = v;
}

__global__ void pack_bias(const float* __restrict__ bi, const float* __restrict__ bf,
                          const float* __restrict__ bc, const float* __restrict__ bo,
                          float* __restrict__ bias) {
  int tid = blockIdx.x * blockDim.x + threadIdx.x;
  if (tid >= NCOL) return;
  int g = tid >> 9;
  int n = tid & (H_ - 1);
  const float* bg = (g == 0) ? bi : (g == 1) ? bf : (g == 2) ? bc : bo;
  bias[tid] = bg[n];
}

// One workgroup per 16 batch rows; full hidden state + full time loop inside.
__global__ __launch_bounds__(1024, 1)
void lstm_all(const float* __restrict__ x,     // (B, T, F)
              const float* __restrict__ Wpk,   // packed (384, 2048, 2)
              const float* __restrict__ bias,  // (2048,)
              float* __restrict__ out) {       // (B, T, H)
  // Padded h buffer: stride 516 dwords -> conflict-free strided v2f reads.
  __shared__ float hbuf[16][516];

  const int tid   = threadIdx.x;
  const int wave  = tid >> 5;        // 0..31 -> 16-col tile of H
  const int lane  = tid & 31;
  const int row   = lane & 15;       // A-frag: M index; B/C/D-frag: N index
  const int hi    = lane >> 4;
  const int khalf = hi << 1;         // A/B frag K offset per ISA layout
  const int nH    = (wave << 4) + row;  // column within H (0..511)
  const int mb    = blockIdx.x << 4;    // batch row base

  // h0 = 0
  for (int i = tid; i < 16 * 516; i += 1024) (&hbuf[0][0])[i] = 0.0f;
  __syncthreads();

  // Per-lane gate biases and packed-layout column offsets (floats)
  const float bI = bias[nH];
  const float bF = bias[H_ + nH];
  const float bC = bias[2 * H_ + nH];
  const float bO = bias[3 * H_ + nH];
  const size_t p0 = (size_t)nH * 2;               // gate i
  const size_t p1 = (size_t)(H_ + nH) * 2;        // gate f
  const size_t p2 = (size_t)(2 * H_ + nH) * 2;    // gate c
  const size_t p3 = (size_t)(3 * H_ + nH) * 2;    // gate o

  const float* xrow = x + (size_t)(mb + row) * (T_ * F_);

  v8f cst = {0.f, 0.f, 0.f, 0.f, 0.f, 0.f, 0.f, 0.f};  // cell state (VGPRs)

  for (int t = 0; t < T_; ++t) {
    v8f a0 = {0.f, 0.f, 0.f, 0.f, 0.f, 0.f, 0.f, 0.f};
    v8f a1 = a0, a2 = a0, a3 = a0;

    const float* xr = xrow + (size_t)t * F_;

    // ---- K = 0..255 : x_t @ W --------------------------------------------
    #pragma unroll 8
    for (int k0 = 0; k0 < F_; k0 += 4) {
      const int k = k0 + khalf;                       // even
      v2f av = *(const v2f*)(xr + k);
      const float* wb = Wpk + (size_t)(k >> 1) * (2 * NCOL);
      v2f b0 = *(const v2f*)(wb + p0);
      v2f b1 = *(const v2f*)(wb + p1);
      v2f b2 = *(const v2f*)(wb + p2);
      v2f b3 = *(const v2f*)(wb + p3);
      a0 = wmma_f32(av, b0, a0);
      a1 = wmma_f32(av, b1, a1);
      a2 = wmma_f32(av, b2, a2);
      a3 = wmma_f32(av, b3, a3);
    }

    // ---- K = 256..767 : h_prev @ U  (h from LDS) -------------------------
    #pragma unroll 8
    for (int k0 = 0; k0 < H_; k0 += 4) {
      const int k = k0 + khalf;                       // even
      v2f av = *(const v2f*)(&hbuf[row][k]);
      const float* wb = Wpk + (size_t)((F_ + k) >> 1) * (2 * NCOL);
      v2f b0 = *(const v2f*)(wb + p0);
      v2f b1 = *(const v2f*)(wb + p1);
      v2f b2 = *(const v2f*)(wb + p2);
      v2f b3 = *(const v2f*)(wb + p3);
      a0 = wmma_f32(av, b0, a0);
      a1 = wmma_f32(av, b1, a1);
      a2 = wmma_f32(av, b2, a2);
      a3 = wmma_f32(av, b3, a3);
    }

    __syncthreads();  // everyone is done reading hbuf for this step

    // ---- gates + state update, in-register (C/D layout: M = v + 8*hi) ---
    #pragma unroll
    for (int v = 0; v < 8; ++v) {
      const int ml = v + (hi << 3);
      float gi = sigmoidf_(a0[v] + bI);
      float gf = sigmoidf_(a1[v] + bF);
      float gc = tanhf(a2[v] + bC);
      float go = sigmoidf_(a3[v] + bO);
      float cn = gf * cst[v] + gi * gc;
      cst[v]   = cn;
      float hn = go * tanhf(cn);
      hbuf[ml][nH] = hn;
      out[(size_t)(mb + ml) * ((size_t)T_ * H_) + (size_t)t * H_ + nH] = hn;
    }

    __syncthreads();  // h_new visible before next step reads it
  }
}

extern "C" void kernel_launch(void* const* d_in, const int* in_sizes, int n_in,
                              void* d_out, int out_size, void* d_ws, size_t ws_size,
                              hipStream_t stream) {
  (void)in_sizes; (void)n_in; (void)out_size; (void)ws_size;
  const float* x  = (const float*)d_in[0];
  const float* Wi = (const float*)d_in[1];
  const float* Ui = (const float*)d_in[2];
  const float* bi = (const float*)d_in[3];
  const float* Wf = (const float*)d_in[4];
  const float* Uf = (const float*)d_in[5];
  const float* bf = (const float*)d_in[6];
  const float* Wc = (const float*)d_in[7];
  const float* Uc = (const float*)d_in[8];
  const float* bc = (const float*)d_in[9];
  const float* Wo = (const float*)d_in[10];
  const float* Uo = (const float*)d_in[11];
  const float* bo = (const float*)d_in[12];

  float* ws   = (float*)d_ws;           // needs ~6.3 MB
  float* Wpk  = ws;
  float* bias = ws + WC_FLOATS;

  pack_weights<<<(WC_FLOATS + 255) / 256, 256, 0, stream>>>(
      Wi, Ui, Wf, Uf, Wc, Uc, Wo, Uo, Wpk);
  pack_bias<<<(NCOL + 255) / 256, 256, 0, stream>>>(bi, bf, bc, bo, bias);

  lstm_all<<<dim3(B_ / 16), dim3(1024), 0, stream>>>(x, Wpk, bias,
                                                     (float*)d_out);
}